// YOLOV1LossV1_59124519797020
// MI455X (gfx1250) — compile-verified
//
#include <hip/hip_runtime.h>
#include <hip/hip_bf16.h>
#include <cstddef>

namespace {

constexpr int   kS   = 7;
constexpr int   kCH  = 30;      // B*5 + C = 2*5 + 20
constexpr float kIMG = 448.0f;
constexpr float kGammaCoord = 5.0f;
constexpr float kGammaNoobj = 0.5f;
constexpr int   kBlocks  = 512;
constexpr int   kThreads = 256;

typedef __attribute__((ext_vector_type(2))) float v2f;
typedef __attribute__((ext_vector_type(8))) float v8f;

// ---- exact f32 wave reduction using the matrix pipe --------------------
// D = A(16x4) * B(4x16) + 0 with A == all-ones sums the 4 K-entries of every
// B column. B's two per-lane VGPRs hold {p0,p1}, so the 4x16 B tile contains
// all 64 per-lane partials of the wave; every row of D equals the 16 column
// sums (exact: all products are 1.0f * x). Four shfl_xor steps finish.
__device__ __forceinline__ float wave_sum(float p0, float p1) {
#if __has_builtin(__builtin_amdgcn_wmma_f32_16x16x4_f32)
  v2f a; a[0] = 1.0f; a[1] = 1.0f;
  v2f b; b[0] = p0;  b[1] = p1;
  v8f c = {};
  c = __builtin_amdgcn_wmma_f32_16x16x4_f32(false, a, false, b, (short)0, c,
                                            false, false);
  float s = c[0];                 // lane L holds colsum[L % 16]
  s += __shfl_xor(s, 1, 32);
  s += __shfl_xor(s, 2, 32);
  s += __shfl_xor(s, 4, 32);
  s += __shfl_xor(s, 8, 32);
  return s;                       // every lane: full wave total
#else
  float s = p0 + p1;
  for (int m = 16; m; m >>= 1) s += __shfl_xor(s, m, 32);
  return s;
#endif
}

__device__ __forceinline__ float block_sum(float p0, float p1) {
  __shared__ float lsum[kThreads / 32];
  const float w = wave_sum(p0, p1);
  const int lane = threadIdx.x & 31;
  const int wid  = threadIdx.x >> 5;
  if (lane == 0) lsum[wid] = w;
  __syncthreads();
  float tot = 0.0f;
  if (threadIdx.x == 0) {
#pragma unroll
    for (int k = 0; k < kThreads / 32; ++k) tot += lsum[k];
  }
  return tot;   // valid in thread 0 only
}

// ---- per-cell YOLOv1 loss ----------------------------------------------
__device__ __forceinline__ float cell_loss(const float* __restrict__ out,
                                           const float* __restrict__ tgt,
                                           const int* __restrict__ gm,
                                           int cell) {
  const int   ij    = cell % (kS * kS);
  const float cellf = 1.0f / (float)kS;
  const float gx = (float)(ij % kS) * cellf;   // arange over axis 2 (j)
  const float gy = (float)(ij / kS) * cellf;   // arange over axis 1 (i)

  // 120-byte record, 8B aligned -> 15 x b64 loads per tensor, coalesced
  const float2* po = (const float2*)(out + (size_t)cell * kCH);
  const float2* pt = (const float2*)(tgt + (size_t)cell * kCH);
  float2 ov[15], tv[15];
#pragma unroll
  for (int k = 0; k < 15; ++k) { ov[k] = po[k]; tv[k] = pt[k]; }

  const float obj = (gm[cell] != 0) ? 1.0f : 0.0f;

  // raw boxes: box0 = [x y w h c] at 0..4, box1 at 5..9
  const float o_x0 = ov[0].x, o_y0 = ov[0].y, o_w0 = ov[1].x, o_h0 = ov[1].y, o_c0 = ov[2].x;
  const float o_x1 = ov[2].y, o_y1 = ov[3].x, o_w1 = ov[3].y, o_h1 = ov[4].x, o_c1 = ov[4].y;
  const float t_x0 = tv[0].x, t_y0 = tv[0].y, t_w0 = tv[1].x, t_h0 = tv[1].y, t_c0 = tv[2].x;
  const float t_c1 = tv[4].y;

  // decoded target box 0 corners (reference slices [..., :1])
  const float tcx = (t_x0 * cellf + gx) * kIMG;
  const float tcy = (t_y0 * cellf + gy) * kIMG;
  const float twp = t_w0 * kIMG, thp = t_h0 * kIMG;
  const float tx1 = tcx - twp * 0.5f, tx2 = tcx + twp * 0.5f;
  const float ty1 = tcy - thp * 0.5f, ty2 = tcy + thp * 0.5f;
  const float area_t = (tx2 - tx1) * (ty2 - ty1);

  auto iou = [&](float bx, float by, float bw, float bh) -> float {
    const float cx = (bx * cellf + gx) * kIMG;
    const float cy = (by * cellf + gy) * kIMG;
    const float w = bw * kIMG, h = bh * kIMG;
    const float x1 = cx - w * 0.5f, x2 = cx + w * 0.5f;
    const float y1 = cy - h * 0.5f, y2 = cy + h * 0.5f;
    const float iw = fmaxf(fminf(x2, tx2) - fmaxf(x1, tx1), 0.0f);
    const float ih = fmaxf(fminf(y2, ty2) - fmaxf(y1, ty1), 0.0f);
    const float inter  = iw * ih;
    const float area_o = (x2 - x1) * (y2 - y1);
    return inter / (area_o + area_t - inter);
  };

  const float iou0 = iou(o_x0, o_y0, o_w0, o_h0);
  const float iou1 = iou(o_x1, o_y1, o_w1, o_h1);
  const int   mid  = (iou1 > iou0) ? 1 : 0;   // jnp.argmax: first max wins
  const float miou = fmaxf(iou0, iou1);

  // box_id_reg == mid whenever obj != 0; otherwise the terms are zeroed
  const float so_x = mid ? o_x1 : o_x0, so_y = mid ? o_y1 : o_y0;
  const float so_w = mid ? o_w1 : o_w0, so_h = mid ? o_h1 : o_h0;
  const float st_x = mid ? tv[2].y : t_x0, st_y = mid ? tv[3].x : t_y0;
  const float st_w = mid ? tv[3].y : t_w0, st_h = mid ? tv[4].x : t_h0;

  const float dx = so_x - st_x, dy = so_y - st_y;
  const float xy = dx * dx + dy * dy;
  const float dw = sqrtf(so_w) - sqrtf(st_w);
  const float dh = sqrtf(so_h) - sqrtf(st_h);
  const float wh = dw * dw + dh * dh;

  // box_id_conf: argmax(onehot*miou*obj) -> mid if miou>0 else 0;
  // sel_iou collapses to miou whenever obj != 0 (0 == miou when miou == 0)
  const int   cid   = (miou > 0.0f) ? mid : 0;
  const float sel_c = cid ? o_c1 : o_c0;
  const float dcf   = sel_c - miou;
  const float cobj  = dcf * dcf;

  const float dn0 = o_c0 - t_c0, dn1 = o_c1 - t_c1;
  const float cnoobj = dn0 * dn0 + dn1 * dn1;

  float cls = 0.0f;
#pragma unroll
  for (int k = 5; k < 15; ++k) {
    const float ax = ov[k].x - tv[k].x;
    const float ay = ov[k].y - tv[k].y;
    cls += ax * ax + ay * ay;
  }

  return obj * (kGammaCoord * (xy + wh) + cobj + cls) +
         (1.0f - obj) * kGammaNoobj * cnoobj;
}

__global__ void __launch_bounds__(kThreads)
yolo_partials(const float* __restrict__ out, const float* __restrict__ tgt,
              const int* __restrict__ gm, float* __restrict__ partials,
              int ncells) {
  float p0 = 0.0f, p1 = 0.0f;
  const int stride = gridDim.x * blockDim.x;
  int flip = 0;
  for (int i = blockIdx.x * blockDim.x + threadIdx.x; i < ncells; i += stride) {
    if (i + stride < ncells) {  // stream-ahead: lowers to global_prefetch_b8
      __builtin_prefetch(out + (size_t)(i + stride) * kCH, 0, 1);
      __builtin_prefetch(tgt + (size_t)(i + stride) * kCH, 0, 1);
    }
    const float v = cell_loss(out, tgt, gm, i);
    if (flip) p1 += v; else p0 += v;   // 2 accumulators feed the WMMA B tile
    flip ^= 1;
  }
  const float tot = block_sum(p0, p1);
  if (threadIdx.x == 0) partials[blockIdx.x] = tot;
}

__global__ void __launch_bounds__(kThreads)
yolo_finalize(const float* __restrict__ partials, float* __restrict__ d_out,
              int n, float inv_bs) {
  float p0 = 0.0f, p1 = 0.0f;
  for (int k = threadIdx.x; k < n; k += 2 * kThreads) p0 += partials[k];
  for (int k = threadIdx.x + kThreads; k < n; k += 2 * kThreads) p1 += partials[k];
  const float tot = block_sum(p0, p1);
  if (threadIdx.x == 0) d_out[0] = tot * inv_bs;
}

}  // namespace

extern "C" void kernel_launch(void* const* d_in, const int* in_sizes, int n_in,
                              void* d_out, int out_size, void* d_ws,
                              size_t ws_size, hipStream_t stream) {
  (void)n_in; (void)out_size; (void)ws_size;
  const float* out = (const float*)d_in[0];
  const float* tgt = (const float*)d_in[1];
  const int*   gm  = (const int*)d_in[2];
  const int    ncells = in_sizes[2];                         // BS * S * S
  const float  inv_bs = (float)(kS * kS) / (float)ncells;    // == 1 / BS
  float* partials = (float*)d_ws;                            // kBlocks floats

  yolo_partials<<<kBlocks, kThreads, 0, stream>>>(out, tgt, gm, partials, ncells);
  yolo_finalize<<<1, kThreads, 0, stream>>>(partials, (float*)d_out, kBlocks, inv_bs);
}